// HeteroLightGCN_13967233647627
// MI455X (gfx1250) — compile-verified
//
#include <hip/hip_runtime.h>

// Problem constants (match reference)
#define NU_ 160000
#define NM_ 60000
#define DD_ 64
#define EE_ 5000000
#define LL_ 200000
#define COMB_W 1.0833333333333333f   // 1/2 + 1/3 + 1/4

typedef __attribute__((ext_vector_type(2))) float v2f;
typedef __attribute__((ext_vector_type(8))) float v8f;

// ---------------- degree / normalization ----------------

__global__ void zero_deg_kernel(unsigned int* deg, int n) {
    int i = blockIdx.x * blockDim.x + threadIdx.x;
    if (i < n) deg[i] = 0u;
}

__global__ void degree_kernel(const int* __restrict__ ef, const int* __restrict__ et,
                              unsigned int* __restrict__ deg_m, unsigned int* __restrict__ deg_u) {
    int e = blockIdx.x * blockDim.x + threadIdx.x;
    if (e < EE_) {
        atomicAdd(&deg_m[ef[e]], 1u);
        atomicAdd(&deg_u[et[e]], 1u);
    }
}

__global__ void inv_sqrt_kernel(const unsigned int* __restrict__ deg, float* __restrict__ inv, int n) {
    int i = blockIdx.x * blockDim.x + threadIdx.x;
    if (i < n) {
        unsigned int d = deg[i];
        inv[i] = d ? rsqrtf((float)d) : 0.0f;
    }
}

// ---------------- res := x (identity node_id gather collapses to copy) ----------------

__global__ void init_res_kernel(const float4* __restrict__ xu, const float4* __restrict__ xm,
                                float4* __restrict__ ru, float4* __restrict__ rm) {
    int i = blockIdx.x * blockDim.x + threadIdx.x;
    const int nu4 = NU_ * DD_ / 4;
    const int nm4 = NM_ * DD_ / 4;
    if (i < nu4)            ru[i] = xu[i];
    else if (i < nu4 + nm4) rm[i - nu4] = xm[i - nu4];
}

// ---------------- single fused edge pass (3 layers collapse to one) ----------------
// One wave32 per edge; lane L handles feature dims L and L+32 (two 128B coalesced streams).

__global__ void edge_agg_kernel(const float* __restrict__ xu, const float* __restrict__ xm,
                                const int* __restrict__ ef, const int* __restrict__ et,
                                const float* __restrict__ inv_m, const float* __restrict__ inv_u,
                                float* __restrict__ res_u, float* __restrict__ res_m) {
    int wave = (blockIdx.x * blockDim.x + threadIdx.x) >> 5;   // wave-uniform
    int lane = threadIdx.x & 31;
    if (wave >= EE_) return;

    int f = ef[wave];
    int t = et[wave];
    float c = COMB_W * inv_m[f] * inv_u[t];

    const float* xmr = xm + (size_t)f * DD_;
    const float* xur = xu + (size_t)t * DD_;
    float* ru = res_u + (size_t)t * DD_;
    float* rm = res_m + (size_t)f * DD_;

    float m0 = xmr[lane], m1 = xmr[lane + 32];
    float u0 = xur[lane], u1 = xur[lane + 32];

    unsafeAtomicAdd(&ru[lane],      c * m0);
    unsafeAtomicAdd(&ru[lane + 32], c * m1);
    unsafeAtomicAdd(&rm[lane],      c * u0);
    unsafeAtomicAdd(&rm[lane + 32], c * u1);
}

// ---------------- classifier: 16 label-pairs per wave via V_WMMA_F32_16X16X4_F32 ----------------
// C[m][n] = dot(res_u[lu[m]], res_m[lm[n]]); scores are the diagonal.
// A(16x4 f32) lane layout: lanes 0-15 K={0,1}, lanes 16-31 K={2,3}; B(4x16) mirrors it,
// so each lane issues one aligned float2 load per matrix per k-step.

__global__ void classifier_wmma_kernel(const float* __restrict__ ru, const float* __restrict__ rm,
                                       const int* __restrict__ lu, const int* __restrict__ lm,
                                       float* __restrict__ scores) {
    __shared__ float cbuf[8 * 256];   // one 16x16 f32 tile per wave (8 waves/block)

    int lane = threadIdx.x & 31;
    int wib  = threadIdx.x >> 5;
    int tile = blockIdx.x * 8 + wib;
    if (tile >= LL_ / 16) return;     // wave-uniform: surviving waves keep EXEC all-ones

    int r    = lane & 15;             // matrix row (A) / col (B) this lane services
    int hi   = lane >> 4;             // 0: K pair {0,1}; 1: K pair {2,3}
    int base = tile * 16;

    const float* up = ru + (size_t)lu[base + r] * DD_;
    const float* mp = rm + (size_t)lm[base + r] * DD_;

    v8f c = {};
#pragma unroll
    for (int k = 0; k < 16; ++k) {
        v2f a = *(const v2f*)(up + 4 * k + 2 * hi);
        v2f b = *(const v2f*)(mp + 4 * k + 2 * hi);
        c = __builtin_amdgcn_wmma_f32_16x16x4_f32(
                /*neg_a=*/false, a, /*neg_b=*/false, b,
                /*c_mod=*/(short)0, c, /*reuse_a=*/false, /*reuse_b=*/false);
    }

    // Spill C tile to this wave's LDS slab per documented layout, pull the diagonal.
    float* w = cbuf + wib * 256;
#pragma unroll
    for (int j = 0; j < 8; ++j) {
        int M = j + hi * 8;           // lanes 0-15: M=j ; lanes 16-31: M=j+8
        w[M * 16 + r] = c[j];
    }
    if (lane < 16) scores[base + lane] = w[lane * 16 + lane];
}

// ---------------- launch ----------------

extern "C" void kernel_launch(void* const* d_in, const int* in_sizes, int n_in,
                              void* d_out, int out_size, void* d_ws, size_t ws_size,
                              hipStream_t stream) {
    const float* emb_user  = (const float*)d_in[0];   // [NU, 64]
    const float* emb_movie = (const float*)d_in[1];   // [NM, 64]
    // d_in[2], d_in[3]: node_id aranges -> identity gather, unused
    const int* edge_from   = (const int*)d_in[4];     // [E] movie ids
    const int* edge_to     = (const int*)d_in[5];     // [E] user ids
    const int* label_movie = (const int*)d_in[6];     // [L]
    const int* label_user  = (const int*)d_in[7];     // [L]

    float* scores = (float*)d_out;                    // [L]
    float* res_u  = scores + LL_;                     // [NU*64]
    float* res_m  = res_u + (size_t)NU_ * DD_;        // [NM*64]

    unsigned int* deg_m = (unsigned int*)d_ws;        // [NM]
    unsigned int* deg_u = deg_m + NM_;                // [NU]
    float* inv_m = (float*)(deg_u + NU_);             // [NM]
    float* inv_u = inv_m + NM_;                       // [NU]

    const int T = 256;
    int ndeg = NM_ + NU_;

    zero_deg_kernel<<<(ndeg + T - 1) / T, T, 0, stream>>>(deg_m, ndeg);
    degree_kernel<<<(EE_ + T - 1) / T, T, 0, stream>>>(edge_from, edge_to, deg_m, deg_u);
    inv_sqrt_kernel<<<(ndeg + T - 1) / T, T, 0, stream>>>(deg_m, inv_m, ndeg);

    int ncopy4 = (NU_ * DD_ + NM_ * DD_) / 4;
    init_res_kernel<<<(ncopy4 + T - 1) / T, T, 0, stream>>>(
        (const float4*)emb_user, (const float4*)emb_movie, (float4*)res_u, (float4*)res_m);

    long long edge_threads = (long long)EE_ * 32;
    edge_agg_kernel<<<(unsigned)((edge_threads + T - 1) / T), T, 0, stream>>>(
        emb_user, emb_movie, edge_from, edge_to, inv_m, inv_u, res_u, res_m);

    int ntiles = LL_ / 16;                            // 12500, exact
    classifier_wmma_kernel<<<(ntiles + 7) / 8, T, 0, stream>>>(
        res_u, res_m, label_user, label_movie, scores);
}